// SoftArgmaxL2Loss_33930241638874
// MI455X (gfx1250) — compile-verified
//
#include <hip/hip_runtime.h>
#include <math.h>
#include <stdint.h>

// Problem constants (match reference setup: B=4, N=256, H=W=256)
#define HW        65536        // H*W per row
#define NTILES    16           // HW / (256 threads * 16 elems)
#define CHUNK     16           // floats per thread per tile
#define BUF_FLTS  4096         // floats per LDS buffer (256 threads * 16)
#define SMEM_FLTS 8192         // double buffer

// (1/tau) * log2(e): softmax_e(x/tau) == softmax_2(x * C)
#define CSCALE 72.13475204444817f

__device__ __forceinline__ void async_issue_tile(unsigned lds_addr, unsigned goff,
                                                 unsigned long long base)
{
    // 4 x b128 per lane = 64 bytes/lane = one 16-float chunk.
    // INST_OFFSET is added to BOTH the global and the LDS address (ISA 8.2/4.4),
    // so one (lds_addr, goff) pair covers all four issues.
    // s_wait_dscnt 0 guards buffer reuse: prior ds_loads of this region must
    // land in VGPRs before the async engine may overwrite the LDS (async vs DS
    // ops are mutually unordered).
    asm volatile(
        "s_wait_dscnt 0\n\t"
        "global_load_async_to_lds_b128 %0, %1, %2 offset:0\n\t"
        "global_load_async_to_lds_b128 %0, %1, %2 offset:16\n\t"
        "global_load_async_to_lds_b128 %0, %1, %2 offset:32\n\t"
        "global_load_async_to_lds_b128 %0, %1, %2 offset:48"
        :
        : "v"(lds_addr), "v"(goff), "s"(base)
        : "memory");
}

__device__ __forceinline__ void merge_tuple(float& m, float& s, float& sx, float& sy,
                                            float m2, float s2, float sx2, float sy2)
{
    float mn = fmaxf(m, m2);
    float r1 = __builtin_amdgcn_exp2f(m - mn);
    float r2 = __builtin_amdgcn_exp2f(m2 - mn);
    s  = s  * r1 + s2  * r2;
    sx = sx * r1 + sx2 * r2;
    sy = sy * r1 + sy2 * r2;
    m  = mn;
}

__global__ __launch_bounds__(256) void softargmax_row_kernel(
    const float* __restrict__ logits,
    const float* __restrict__ gt_xy,
    const int*   __restrict__ Himg_p,
    const int*   __restrict__ Wimg_p,
    float*       __restrict__ errs)
{
    __shared__ float smem[SMEM_FLTS];  // 32 KB: two 16 KB tile buffers
    __shared__ float red[32];          // 8 waves x (m,s,sx,sy)

    const int t = threadIdx.x;
    const int r = blockIdx.x;  // row = b*N + n; logits rows are contiguous

    const unsigned long long base =
        (unsigned long long)(uintptr_t)(logits + (size_t)r * HW);

    // LDS byte address of this thread's chunk in buffer 0 (low 32 bits of the
    // flat shared address == LDS offset per ISA aperture rules).
    const unsigned lds0      = (unsigned)(uintptr_t)(&smem[t * CHUNK]);
    const unsigned ldsStride = BUF_FLTS * 4;  // bytes between the two buffers

    // Per-thread coordinate bases: element index l = tile*4096 + t*16 + j
    //   x = l & 255 = (t&15)*16 + j   (chunk never crosses a W-row)
    //   y = l >> 8  = tile*16 + (t>>4)
    const float x0    = (float)((t & 15) * CHUNK);
    const int   ybase = t >> 4;

    // Prologue: fill both buffers.
    async_issue_tile(lds0,             (unsigned)(t * 64),         base);
    async_issue_tile(lds0 + ldsStride, (unsigned)(16384 + t * 64), base);

    float m = -1.0e30f, s = 0.f, sx = 0.f, sy = 0.f;

    for (int tile = 0; tile < NTILES; ++tile) {
        // Outstanding async ops: 4 for this tile (+4 for next, if any).
        if (tile < NTILES - 1)
            asm volatile("s_wait_asynccnt 4" ::: "memory");
        else
            asm volatile("s_wait_asynccnt 0" ::: "memory");

        const float* buf = &smem[(tile & 1) * BUF_FLTS + t * CHUNK];
        float4 a  = ((const float4*)buf)[0];
        float4 b4 = ((const float4*)buf)[1];
        float4 c4 = ((const float4*)buf)[2];
        float4 d4 = ((const float4*)buf)[3];
        float tv[CHUNK] = {a.x,  a.y,  a.z,  a.w,  b4.x, b4.y, b4.z, b4.w,
                           c4.x, c4.y, c4.z, c4.w, d4.x, d4.y, d4.z, d4.w};
#pragma unroll
        for (int j = 0; j < CHUNK; ++j) tv[j] *= CSCALE;

        float cmax = tv[0];
#pragma unroll
        for (int j = 1; j < CHUNK; ++j) cmax = fmaxf(cmax, tv[j]);

        float mn  = fmaxf(m, cmax);
        float rsc = __builtin_amdgcn_exp2f(m - mn);  // rescale old accumulators

        float q = 0.f, qj = 0.f;
#pragma unroll
        for (int j = 0; j < CHUNK; ++j) {
            float p = __builtin_amdgcn_exp2f(tv[j] - mn);
            q += p;
            qj = fmaf(p, (float)j, qj);
        }

        const float yc = (float)(tile * 16 + ybase);
        s  = fmaf(s,  rsc, q);
        sx = fmaf(sx, rsc, fmaf(q, x0, qj));
        sy = fmaf(sy, rsc, q * yc);
        m  = mn;

        // Refill this buffer with tile+2.
        if (tile + 2 < NTILES) {
            unsigned goff = (unsigned)((tile + 2) * 16384 + t * 64);
            async_issue_tile(lds0 + (tile & 1) * ldsStride, goff, base);
        }
    }

    // Wave32 reduction of (m, s, sx, sy).
#pragma unroll
    for (int o = 16; o > 0; o >>= 1) {
        float m2  = __shfl_xor(m,  o, 32);
        float s2  = __shfl_xor(s,  o, 32);
        float sx2 = __shfl_xor(sx, o, 32);
        float sy2 = __shfl_xor(sy, o, 32);
        merge_tuple(m, s, sx, sy, m2, s2, sx2, sy2);
    }

    const int wave = t >> 5, lane = t & 31;
    if (lane == 0) {
        red[wave * 4 + 0] = m;
        red[wave * 4 + 1] = s;
        red[wave * 4 + 2] = sx;
        red[wave * 4 + 3] = sy;
    }
    __syncthreads();

    if (wave == 0) {
        float M = -1.0e30f, S = 0.f, SX = 0.f, SY = 0.f;
        if (lane < 8) {
            M  = red[lane * 4 + 0];
            S  = red[lane * 4 + 1];
            SX = red[lane * 4 + 2];
            SY = red[lane * 4 + 3];
        }
#pragma unroll
        for (int o = 4; o > 0; o >>= 1) {
            float m2  = __shfl_xor(M,  o, 32);
            float s2  = __shfl_xor(S,  o, 32);
            float sx2 = __shfl_xor(SX, o, 32);
            float sy2 = __shfl_xor(SY, o, 32);
            merge_tuple(M, S, SX, SY, m2, s2, sx2, sy2);
        }
        if (lane == 0) {
            float inv = 1.0f / S;
            float px = SX * inv, py = SY * inv;
            // pred_norm = 2*pred/(W-1) - 1, W-1 = H-1 = 255
            float pnx = 2.0f * px * (1.0f / 255.0f) - 1.0f;
            float pny = 2.0f * py * (1.0f / 255.0f) - 1.0f;
            // gt_norm = 2*gt/(img-1) - 1 (scale-in and scale-out cancel)
            float Wimg = (float)(*Wimg_p);
            float Himg = (float)(*Himg_p);
            float gnx = 2.0f * gt_xy[2 * r + 0] / (Wimg - 1.0f) - 1.0f;
            float gny = 2.0f * gt_xy[2 * r + 1] / (Himg - 1.0f) - 1.0f;
            float dx = pnx - gnx, dy = pny - gny;
            errs[r] = sqrtf(dx * dx + dy * dy);
        }
    }
}

__global__ __launch_bounds__(256) void sum_rows_kernel(const float* __restrict__ errs,
                                                       float* __restrict__ out, int n)
{
    __shared__ float sm[8];
    float v = 0.f;
    for (int i = threadIdx.x; i < n; i += 256) v += errs[i];
#pragma unroll
    for (int o = 16; o > 0; o >>= 1) v += __shfl_xor(v, o, 32);
    if ((threadIdx.x & 31) == 0) sm[threadIdx.x >> 5] = v;
    __syncthreads();
    if (threadIdx.x == 0) {
        float a = 0.f;
        for (int i = 0; i < 8; ++i) a += sm[i];
        out[0] = a;
    }
}

extern "C" void kernel_launch(void* const* d_in, const int* in_sizes, int n_in,
                              void* d_out, int out_size, void* d_ws, size_t ws_size,
                              hipStream_t stream)
{
    const float* logits = (const float*)d_in[0];  // (B,N,H,W) f32
    const float* gt_xy  = (const float*)d_in[1];  // (B,N,2)   f32
    // d_in[2] = grid: unused (x,y derived from flat index)
    const int* Himg = (const int*)d_in[3];
    const int* Wimg = (const int*)d_in[4];

    float* out  = (float*)d_out;
    float* errs = (float*)d_ws;  // 1024 floats of scratch

    const int rows = in_sizes[1] / 2;  // B*N = 1024

    softargmax_row_kernel<<<rows, 256, 0, stream>>>(logits, gt_xy, Himg, Wimg, errs);
    sum_rows_kernel<<<1, 256, 0, stream>>>(errs, out, rows);
}